// MetalLLMQwenAttnProcessor_7430293422908
// MI455X (gfx1250) — compile-verified
//
#include <hip/hip_runtime.h>
#include <hip/hip_bf16.h>
#include <math.h>

// ---------------------------------------------------------------------------
// Decode-step attention (Qwen-style), MI455X / gfx1250, wave32.
//   B=8, Q=1, D=4096, NH=32, NKV=8, HD=128, KV=4096, rope base 1e4.
// Memory-bound: ~1.07GB KV stream dominates. Flash-decode split kernel for
// attention; WMMA f32 16x16x4 for the four projection GEMMs (M=8 -> pad 16).
// ---------------------------------------------------------------------------

#define BB   8
#define DD   4096
#define NH   32
#define NKV  8
#define HD   128
#define KVL  4096
#define NQ   (NH*HD)    // 4096
#define NKVD (NKV*HD)   // 1024
#define SPLITS 8
#define CHUNK  (KVL/SPLITS)   // 512
#define WAVES_PER_BLOCK 8
#define KEYS_PER_WAVE (CHUNK/WAVES_PER_BLOCK) // 64

typedef __attribute__((ext_vector_type(2))) float v2f;
typedef __attribute__((ext_vector_type(4))) float v4f;
typedef __attribute__((ext_vector_type(8))) float v8f;

// ---------------------------------------------------------------------------
// GEMM: Y[m][n] = sum_d X[m][d] * W[n][d] + bias[n],  m in [0,8), via WMMA
// f32 16x16x4 with M padded to 16 (rows 8..15 zero-masked).
// A layout (ISA 7.12.2, 32-bit A 16x4): lane = M (0..15) in each half,
//   K = 2*(lane>=16) + vgpr.  B layout assumed symmetric: N = lane&15,
//   K = 2*(lane>=16) + vgpr.  C/D: vgpr r -> M = r + 8*(lane>=16), N = lane&15.
// One wave per 16-column tile; each weight byte read exactly once.
// ---------------------------------------------------------------------------
__global__ void gemm_xwT_wmma(const float* __restrict__ X,   // [8][Dd]
                              const float* __restrict__ W,   // [N][Dd]
                              const float* __restrict__ bias,// [N] or null
                              float* __restrict__ Y,         // [8][N]
                              int N, int Dd)
{
    const int lane = threadIdx.x & 31;
    const int wave = threadIdx.x >> 5;
    const int tile = blockIdx.x * (blockDim.x >> 5) + wave;
    if (tile * 16 >= N) return;                 // whole-wave uniform exit
    const int n0    = tile * 16;
    const int mrow  = lane & 15;
    const int khalf = (lane >> 4) * 2;          // K base for this lane-half

    const float amask = (mrow < 8) ? 1.0f : 0.0f;
    const float* xptr = X + (size_t)(mrow & 7) * Dd + khalf;
    const float* wptr = W + (size_t)(n0 + (lane & 15)) * Dd + khalf;

    v8f c = {};
#pragma unroll 8
    for (int d0 = 0; d0 < Dd; d0 += 4) {
        v2f a = *(const v2f*)(xptr + d0);
        a = a * amask;
        v2f b = *(const v2f*)(wptr + d0);
        c = __builtin_amdgcn_wmma_f32_16x16x4_f32(
                /*neg_a=*/false, a, /*neg_b=*/false, b,
                /*c_mod=*/(short)0, c, /*reuse_a=*/false, /*reuse_b=*/false);
    }

    if (lane < 16) {                            // rows 0..7 live in lanes 0..15
        const int n  = n0 + lane;
        const float bv = bias ? bias[n] : 0.0f;
#pragma unroll
        for (int r = 0; r < 8; ++r)
            Y[(size_t)r * N + n] = c[r] + bv;
    }
}

// ---------------------------------------------------------------------------
// RoPE in-place on q (8x32x128) and new-k (8x8x128), single position.
// grid: B*(NH+NKV) blocks, 128 threads (one per dim).
// ---------------------------------------------------------------------------
__global__ void rope_kernel(float* __restrict__ qbuf, float* __restrict__ kbuf,
                            const int* __restrict__ pos_ptr)
{
    const int r = blockIdx.x;
    const int d = threadIdx.x;                  // 0..127
    float* base;
    if (r < BB * NH)  base = qbuf + (size_t)(r / NH) * NQ + (r % NH) * HD;
    else {
        const int rr = r - BB * NH;
        base = kbuf + (size_t)(rr / NKV) * NKVD + (rr % NKV) * HD;
    }
    const float pos = (float)(*pos_ptr);
    const int   i   = d & 63;                   // freq index
    const float inv = __powf(10000.0f, -(float)i / 64.0f);
    const float ang = pos * inv;
    const float cs  = __cosf(ang), sn = __sinf(ang);

    const float x     = base[d];
    const float other = (d < 64) ? -base[d + 64] : base[d - 64];
    __syncthreads();                            // all reads before any write
    base[d] = x * cs + other * sn;
}

// ---------------------------------------------------------------------------
// Flash-decode partial: one block per (b, h, split). 8 waves, each wave does
// 64 keys with online softmax; lane L holds dims [4L,4L+4) (b128 loads, one
// wave load == one 512B K row). Wave 7 of the last split also handles the new
// key at position KV. Cross-wave merge in LDS; partial (m, l, acc) -> ws.
// ---------------------------------------------------------------------------
__global__ void attn_partial(const float* __restrict__ qbuf,
                             const float* __restrict__ kbuf,
                             const float* __restrict__ vbuf,
                             const float* __restrict__ past_k,
                             const float* __restrict__ past_v,
                             float* __restrict__ pm, float* __restrict__ pl,
                             float* __restrict__ pacc)
{
    const int idx  = blockIdx.x;                 // b*NH*SPLITS + h*SPLITS + s
    const int s    = idx % SPLITS;
    const int h    = (idx / SPLITS) % NH;
    const int b    = idx / (SPLITS * NH);
    const int lane = threadIdx.x & 31;
    const int wave = threadIdx.x >> 5;

    const float scale = 0.08838834764831845f;    // 1/sqrt(128)
    v4f qv = *(const v4f*)(qbuf + (size_t)b * NQ + h * HD + lane * 4);
    qv = qv * scale;

    const size_t head_off = (size_t)(b * NH + h) * (size_t)KVL * HD;
    const float* K = past_k + head_off;
    const float* V = past_v + head_off;

    float m = -INFINITY, l = 0.0f;
    v4f acc = {};

    const int kbase = s * CHUNK + wave * KEYS_PER_WAVE;
    for (int kk = 0; kk < KEYS_PER_WAVE; ++kk) {
        const size_t row = (size_t)(kbase + kk) * HD + lane * 4;
        __builtin_prefetch(K + row + 8 * HD, 0, 0);   // global_prefetch_b8
        __builtin_prefetch(V + row + 8 * HD, 0, 0);
        v4f kr = *(const v4f*)(K + row);
        float dot = qv.x * kr.x + qv.y * kr.y + qv.z * kr.z + qv.w * kr.w;
#pragma unroll
        for (int off = 16; off >= 1; off >>= 1)
            dot += __shfl_xor(dot, off, 32);
        const float mnew = fmaxf(m, dot);
        const float corr = __expf(m - mnew);     // exp(-inf)=0 on first iter
        const float p    = __expf(dot - mnew);
        v4f vr = *(const v4f*)(V + row);
        acc = acc * corr + vr * p;
        l   = l * corr + p;
        m   = mnew;
    }

    // new token's key/value (position KV): visible (qpos >= all kpos).
    if (s == SPLITS - 1 && wave == WAVES_PER_BLOCK - 1) {
        const int g = h / (NH / NKV);
        v4f kr = *(const v4f*)(kbuf + (size_t)b * NKVD + g * HD + lane * 4);
        float dot = qv.x * kr.x + qv.y * kr.y + qv.z * kr.z + qv.w * kr.w;
#pragma unroll
        for (int off = 16; off >= 1; off >>= 1)
            dot += __shfl_xor(dot, off, 32);
        const float mnew = fmaxf(m, dot);
        const float corr = __expf(m - mnew);
        const float p    = __expf(dot - mnew);
        v4f vr = *(const v4f*)(vbuf + (size_t)b * NKVD + g * HD + lane * 4);
        acc = acc * corr + vr * p;
        l   = l * corr + p;
        m   = mnew;
    }

    // cross-wave merge in LDS
    __shared__ float sm[WAVES_PER_BLOCK];
    __shared__ float sl[WAVES_PER_BLOCK];
    __shared__ float sacc[WAVES_PER_BLOCK * HD];
    if (lane == 0) { sm[wave] = m; sl[wave] = l; }
    *(v4f*)(sacc + wave * HD + lane * 4) = acc;
    __syncthreads();

    if (wave == 0) {
        float M = -INFINITY;
#pragma unroll
        for (int w = 0; w < WAVES_PER_BLOCK; ++w) M = fmaxf(M, sm[w]);
        float L = 0.0f;
        v4f A = {};
#pragma unroll
        for (int w = 0; w < WAVES_PER_BLOCK; ++w) {
            const float cw = __expf(sm[w] - M);
            L += sl[w] * cw;
            v4f aw = *(const v4f*)(sacc + w * HD + lane * 4);
            A = A + aw * cw;
        }
        if (lane == 0) { pm[idx] = M; pl[idx] = L; }
        *(v4f*)(pacc + (size_t)idx * HD + lane * 4) = A;
    }
}

// ---------------------------------------------------------------------------
// Combine splits: one block per (b,h), 128 threads (one per dim).
// ctx[b][h*HD + d] = sum_s exp(m_s - M) * acc_s[d] / sum_s exp(m_s - M) * l_s
// ---------------------------------------------------------------------------
__global__ void attn_combine(const float* __restrict__ pm,
                             const float* __restrict__ pl,
                             const float* __restrict__ pacc,
                             float* __restrict__ ctx)
{
    const int bh = blockIdx.x;                   // b*NH + h
    const int d  = threadIdx.x;                  // 0..127
    const int p0 = bh * SPLITS;
    float M = -INFINITY;
#pragma unroll
    for (int s = 0; s < SPLITS; ++s) M = fmaxf(M, pm[p0 + s]);
    float L = 0.0f, A = 0.0f;
#pragma unroll
    for (int s = 0; s < SPLITS; ++s) {
        const float c = __expf(pm[p0 + s] - M);
        L += pl[p0 + s] * c;
        A += pacc[(size_t)(p0 + s) * HD + d] * c;
    }
    ctx[(size_t)bh * HD + d] = A / L;            // ctx laid out [b][h*HD+d]
}

// ---------------------------------------------------------------------------
extern "C" void kernel_launch(void* const* d_in, const int* in_sizes, int n_in,
                              void* d_out, int out_size, void* d_ws, size_t ws_size,
                              hipStream_t stream)
{
    const float* hidden = (const float*)d_in[0];   // [8][1][4096]
    const float* past_k = (const float*)d_in[1];   // [8][32][4096][128]
    const float* past_v = (const float*)d_in[2];   // [8][32][4096][128]
    const float* Wq     = (const float*)d_in[3];   // [4096][4096]
    const float* bq     = (const float*)d_in[4];   // [4096]
    const float* Wk     = (const float*)d_in[5];   // [1024][4096]
    const float* bk     = (const float*)d_in[6];   // [1024]
    const float* Wv     = (const float*)d_in[7];   // [1024][4096]
    const float* bv     = (const float*)d_in[8];   // [1024]
    const float* Wo     = (const float*)d_in[9];   // [4096][4096]
    const int*   pos    = (const int*)d_in[10];    // scalar 4096

    float* ws   = (float*)d_ws;
    float* qbuf = ws;                                   // 8*4096
    float* kbuf = qbuf + (size_t)BB * NQ;               // 8*1024
    float* vbuf = kbuf + (size_t)BB * NKVD;             // 8*1024
    float* ctxb = vbuf + (size_t)BB * NKVD;             // 8*4096
    float* pm   = ctxb + (size_t)BB * NQ;               // 2048
    float* pl   = pm + BB * NH * SPLITS;                // 2048
    float* pacc = pl + BB * NH * SPLITS;                // 2048*128

    // 1) projections (WMMA f32 16x16x4): q, k, v
    gemm_xwT_wmma<<<NQ / 16 / WAVES_PER_BLOCK, 256, 0, stream>>>(hidden, Wq, bq, qbuf, NQ, DD);
    gemm_xwT_wmma<<<NKVD / 16 / WAVES_PER_BLOCK, 256, 0, stream>>>(hidden, Wk, bk, kbuf, NKVD, DD);
    gemm_xwT_wmma<<<NKVD / 16 / WAVES_PER_BLOCK, 256, 0, stream>>>(hidden, Wv, bv, vbuf, NKVD, DD);

    // 2) RoPE on q and new k
    rope_kernel<<<BB * (NH + NKV), HD, 0, stream>>>(qbuf, kbuf, pos);

    // 3) flash-decode partials over KV splits (bandwidth-dominant stage)
    attn_partial<<<BB * NH * SPLITS, WAVES_PER_BLOCK * 32, 0, stream>>>(
        qbuf, kbuf, vbuf, past_k, past_v, pm, pl, pacc);

    // 4) combine splits -> ctx
    attn_combine<<<BB * NH, HD, 0, stream>>>(pm, pl, pacc, ctxb);

    // 5) output projection -> d_out
    gemm_xwT_wmma<<<NQ / 16 / WAVES_PER_BLOCK, 256, 0, stream>>>(
        ctxb, Wo, nullptr, (float*)d_out, NQ, DD);
}